// SageEncoder_85873576117016
// MI455X (gfx1250) — compile-verified
//
#include <hip/hip_runtime.h>

#define NODES 100000
#define EDGES 1600000
#define D 64
#define EPS 1e-12f

typedef __attribute__((ext_vector_type(2))) float v2f;
typedef __attribute__((ext_vector_type(8))) float v8f;

// ---------------------------------------------------------------- zero scratch
__global__ void zero_kernel(float4* __restrict__ p, int n4) {
  int i = blockIdx.x * blockDim.x + threadIdx.x;
  if (i < n4) p[i] = make_float4(0.f, 0.f, 0.f, 0.f);
}

// ------------------------------------------------- edge scatter: agg[dst] += x[src]
// 16 threads per edge, each handling a float4 chunk of the 64-float row.
// 16 consecutive lanes hit one contiguous destination row -> cacheline-granular
// atomic packets in L2 (whole working set is L2-resident on MI455X: 192MB L2).
__global__ void scatter_kernel(const float* __restrict__ xin,
                               const int* __restrict__ ei,
                               float* __restrict__ agg) {
  int idx = blockIdx.x * blockDim.x + threadIdx.x;  // EDGES*16 threads exactly
  int e = idx >> 4;
  int q = (idx & 15) << 2;
  int src = ei[e];
  int dst = ei[EDGES + e];
  const float4 v = *(const float4*)(xin + (size_t)src * D + q);
  float* o = agg + (size_t)dst * D + q;
  atomicAdd(o + 0, v.x);
  atomicAdd(o + 1, v.y);
  atomicAdd(o + 2, v.z);
  atomicAdd(o + 3, v.w);
}

// ------------------------------------------------- fused SAGE linear via f32 WMMA
// Out[m, n] = sum_k Agg[m,k]*Wl[n,k] + sum_k X[m,k]*Wr[n,k] + bias[n]  (+ ReLU)
// Block = 128 threads = 4 waves; block computes 16 rows x 64 cols; each wave
// one 16x16 tile via 16 K-steps of V_WMMA_F32_16X16X4_F32 per chain.
__global__ __launch_bounds__(128) void sage_gemm_kernel(
    const float* __restrict__ Agg, const float* __restrict__ X,
    const float* __restrict__ Wl, const float* __restrict__ Wr,
    const float* __restrict__ bias, float* __restrict__ Out,
    int applyRelu) {
  const int lane = threadIdx.x & 31;
  const int wave = threadIdx.x >> 5;
  const int m0 = blockIdx.x << 4;   // 6250 * 16 == 100000 exactly, no tail
  const int n0 = wave << 4;
  const int half = lane >> 4;       // 0: K pair {0,1}; 1: K pair {2,3}
  const int l = lane & 15;

  const float* aRow  = Agg + (size_t)(m0 + l) * D;
  const float* xRow  = X   + (size_t)(m0 + l) * D;
  const float* wlRow = Wl  + (size_t)(n0 + l) * D;  // B[k][n] = Wl[n][k]
  const float* wrRow = Wr  + (size_t)(n0 + l) * D;

  v8f acc0 = {};
  v8f acc1 = {};
#pragma unroll
  for (int kk = 0; kk < 16; ++kk) {
    const int kb = (kk << 2) + (half << 1);
    v2f a0 = *(const v2f*)(aRow + kb);
    v2f b0 = *(const v2f*)(wlRow + kb);
    acc0 = __builtin_amdgcn_wmma_f32_16x16x4_f32(
        false, a0, false, b0, (short)0, acc0, false, false);
    v2f a1 = *(const v2f*)(xRow + kb);
    v2f b1 = *(const v2f*)(wrRow + kb);
    acc1 = __builtin_amdgcn_wmma_f32_16x16x4_f32(
        false, a1, false, b1, (short)0, acc1, false, false);
  }

  // C/D layout: VGPR r -> row m0 + r + 8*half, col n0 + l
  const float bv = bias[n0 + l];
#pragma unroll
  for (int r = 0; r < 8; ++r) {
    float v = acc0[r] + acc1[r] + bv;
    if (applyRelu) v = fmaxf(v, 0.f);  // uniform branch, EXEC stays all-1s
    Out[(size_t)(m0 + r + (half << 3)) * D + n0 + l] = v;
  }
}

// ------------------------------------------------- per-row L2 normalize (in place)
// One wave32 per node; 2 floats per lane; shfl_xor tree reduction.
__global__ void normalize_kernel(float* __restrict__ h) {
  int node = (blockIdx.x * blockDim.x + threadIdx.x) >> 5;  // 12500*256/32 == N
  int lane = threadIdx.x & 31;
  float* row = h + (size_t)node * D;
  float2 v = *(float2*)(row + lane * 2);
  float s = v.x * v.x + v.y * v.y;
#pragma unroll
  for (int off = 16; off > 0; off >>= 1) s += __shfl_xor(s, off, 32);
  float scale = 1.0f / fmaxf(sqrtf(s), EPS);
  v.x *= scale;
  v.y *= scale;
  *(float2*)(row + lane * 2) = v;
}

// ----------------------------------------------------------------------------
extern "C" void kernel_launch(void* const* d_in, const int* in_sizes, int n_in,
                              void* d_out, int out_size, void* d_ws, size_t ws_size,
                              hipStream_t stream) {
  const float* x   = (const float*)d_in[0];
  const int*   ei  = (const int*)d_in[1];
  // d_in[2] = edge_feature: unused by the reference forward
  const float* W1l = (const float*)d_in[3];
  const float* b1l = (const float*)d_in[4];
  const float* W1r = (const float*)d_in[5];
  const float* W2l = (const float*)d_in[6];
  const float* b2l = (const float*)d_in[7];
  const float* W2r = (const float*)d_in[8];
  float* out = (float*)d_out;

  float* agg = (float*)d_ws;                    // 25.6 MB, reused by both layers
  float* h   = agg + (size_t)NODES * D;         // 25.6 MB

  const int n4 = NODES * D / 4;                 // 1.6M float4
  const int zeroBlocks = (n4 + 255) / 256;
  const int scatBlocks = EDGES * 16 / 256;      // 100000
  const int gemmBlocks = NODES / 16;            // 6250
  const int normBlocks = NODES * 32 / 256;      // 12500

  // ---- layer 1
  zero_kernel<<<zeroBlocks, 256, 0, stream>>>((float4*)agg, n4);
  scatter_kernel<<<scatBlocks, 256, 0, stream>>>(x, ei, agg);
  sage_gemm_kernel<<<gemmBlocks, 128, 0, stream>>>(agg, x, W1l, W1r, b1l, h, 1);
  normalize_kernel<<<normBlocks, 256, 0, stream>>>(h);

  // ---- layer 2
  zero_kernel<<<zeroBlocks, 256, 0, stream>>>((float4*)agg, n4);
  scatter_kernel<<<scatBlocks, 256, 0, stream>>>(h, ei, agg);
  sage_gemm_kernel<<<gemmBlocks, 128, 0, stream>>>(agg, h, W2l, W2r, b2l, out, 0);
}